// SequenceFeatureExtractor_28819230556693
// MI455X (gfx1250) — compile-verified
//
#include <hip/hip_runtime.h>
#include <hip/hip_bf16.h>
#include <math.h>

// Problem dims (fixed by reference): S=256, B=16, F=200, H=100
#define SDIM 256
#define BDIM 16
#define FDIM 200
#define HDIM 100

typedef __attribute__((ext_vector_type(2))) float v2f;
typedef __attribute__((ext_vector_type(8))) float v8f;

// ---------------------------------------------------------------------------
// tanh: prefer the CDNA5 hardware transcendental v_tanh_f32 (1 trans op);
// fall back to clamp + exp-based form (v_exp_f32 + fast divide) otherwise.
// ---------------------------------------------------------------------------
__device__ __forceinline__ float fast_tanh(float x) {
#if __has_builtin(__builtin_amdgcn_tanhf)
    return __builtin_amdgcn_tanhf(x);
#else
    float cx = fminf(fmaxf(x, -15.f), 15.f);
    float e  = __expf(2.f * cx);
    return __fdividef(e - 1.f, e + 1.f);
#endif
}

// ---------------------------------------------------------------------------
// Kernel 1: q = X @ Wq ; k' = X @ Wk + bias   (fp32 WMMA 16x16x4)
// X is (S*B=4096) x F=200 row-major; W is F x H; out row-major stride H=100.
// One wave computes one 16x16 output tile.  grid=(256, 7, 2) block=32.
// N-overhang: clamp the B column index (garbage only lands in D columns
// >= H which are never stored) -> branch-free load+wmma steady state.
// ---------------------------------------------------------------------------
__global__ __launch_bounds__(32)
void sfe_proj_kernel(const float* __restrict__ x,
                     const float* __restrict__ Wq,
                     const float* __restrict__ Wk,
                     const float* __restrict__ bias,
                     float* __restrict__ qws,
                     float* __restrict__ kws) {
    const int lane = threadIdx.x & 31;
    const int tm = blockIdx.x;          // M tile (rows of S*B)
    const int tn = blockIdx.y;          // N tile (H)
    const bool isK = (blockIdx.z != 0);
    const float* __restrict__ W   = isK ? Wk : Wq;
    float* __restrict__       dst = isK ? kws : qws;

    const int row  = tm * 16 + (lane & 15);   // A row (M)
    const int kb   = (lane >> 4) * 2;         // K sub-offset per lane half
    const int col  = tn * 16 + (lane & 15);   // B/N column
    const int colc = (col < HDIM) ? col : (HDIM - 1);  // clamped, always valid

    const float* ap = x + row * FDIM + kb;    // A: contiguous pair, 8B aligned
    const float* bp = W + kb * HDIM + colc;   // B: marches by 4*HDIM per step

    v8f acc = {};
    for (int kk = 0; kk < FDIM; kk += 4) {
        v2f a, bv;
        a.x  = ap[0];
        a.y  = ap[1];
        bv.x = bp[0];
        bv.y = bp[HDIM];
        acc = __builtin_amdgcn_wmma_f32_16x16x4_f32(
                false, a, false, bv, (short)0, acc, false, false);
        ap += 4;
        bp += 4 * HDIM;
    }

    if (col < HDIM) {
        const float addb = isK ? bias[col] : 0.f;
        const int rbase = tm * 16 + (lane >> 4) * 8;  // D: rows r / r+8 split
        #pragma unroll
        for (int r = 0; r < 8; ++r)
            dst[(rbase + r) * HDIM + col] = acc[r] + addb;
    }
}

// ---------------------------------------------------------------------------
// Kernel 2: e[t,s,b] = sum_h tanh(q[t,b,h] + k'[s,b,h]) * v[h]  (+ s-mask)
// 16x16 (t,s) tile per block, per batch.  grid=(16,16,16) block=256.
// Dominant cost of the whole problem: 256*256*16*100 ~ 105M tanh evals.
// float2 LDS reads (ds_load_b64) halve DS ops per tanh.
// ---------------------------------------------------------------------------
__global__ __launch_bounds__(256)
void sfe_energy_kernel(const float* __restrict__ qws,
                       const float* __restrict__ kws,
                       const float* __restrict__ vvec,
                       const int*   __restrict__ lens,
                       float* __restrict__ ews) {
    __shared__ float qs[16 * HDIM];
    __shared__ float ks[16 * HDIM];
    __shared__ float vs[HDIM];

    const int tid = threadIdx.x;
    const int tT = blockIdx.x, tS = blockIdx.y, b = blockIdx.z;

    for (int i = tid; i < 16 * HDIM; i += 256) {
        const int r = i / HDIM, c = i % HDIM;
        qs[i] = qws[((tT * 16 + r) * BDIM + b) * HDIM + c];
        ks[i] = kws[((tS * 16 + r) * BDIM + b) * HDIM + c];
    }
    if (tid < HDIM) vs[tid] = vvec[tid];
    __syncthreads();

    const int tt = tid >> 4, ss = tid & 15;
    // rows start at 400B boundaries -> float2 (8B) loads are aligned
    const float2* q2 = reinterpret_cast<const float2*>(qs + tt * HDIM);
    const float2* k2 = reinterpret_cast<const float2*>(ks + ss * HDIM);
    const float2* v2 = reinterpret_cast<const float2*>(vs);

    float acc = 0.f;
    #pragma unroll 5
    for (int i = 0; i < HDIM / 2; ++i) {
        const float2 qv = q2[i];
        const float2 kv = k2[i];
        const float2 vv = v2[i];
        acc += fast_tanh(qv.x + kv.x) * vv.x;
        acc += fast_tanh(qv.y + kv.y) * vv.y;
    }

    const int sg = tS * 16 + ss;
    const float val = (sg < lens[b]) ? acc : -INFINITY;
    ews[(b * SDIM + tT * 16 + tt) * SDIM + sg] = val;
}

// ---------------------------------------------------------------------------
// Kernel 3: softmax over s (in place).  One block per (b,t) row of 256.
// ---------------------------------------------------------------------------
__global__ __launch_bounds__(256)
void sfe_softmax_kernel(float* __restrict__ ews) {
    __shared__ float red[256];
    const int tid = threadIdx.x;
    float* row = ews + (size_t)blockIdx.x * SDIM;

    const float x = row[tid];
    red[tid] = x;
    __syncthreads();
    for (int s = 128; s > 0; s >>= 1) {
        if (tid < s) red[tid] = fmaxf(red[tid], red[tid + s]);
        __syncthreads();
    }
    const float m = red[0];
    __syncthreads();

    const float ex = __expf(x - m);   // -inf -> 0
    red[tid] = ex;
    __syncthreads();
    for (int s = 128; s > 0; s >>= 1) {
        if (tid < s) red[tid] += red[tid + s];
        __syncthreads();
    }
    const float denom = red[0];
    row[tid] = ex / denom;
}

// ---------------------------------------------------------------------------
// Kernel 4: attended[t,b,f] = sum_s a[t,s,b] * x[s,b,f]  (fp32 WMMA 16x16x4)
// Per batch: A(256x256) @ X_b(256x200).  grid=(16,13,16) block=32.
// Same clamped-column trick as kernel 1: no branches in the K loop.
// ---------------------------------------------------------------------------
__global__ __launch_bounds__(32)
void sfe_attend_kernel(const float* __restrict__ aws,
                       const float* __restrict__ x,
                       float* __restrict__ att) {
    const int lane = threadIdx.x & 31;
    const int tm = blockIdx.x, tn = blockIdx.y, b = blockIdx.z;

    const int trow = tm * 16 + (lane & 15);
    const int kb   = (lane >> 4) * 2;
    const int fcol = tn * 16 + (lane & 15);
    const int fc   = (fcol < FDIM) ? fcol : (FDIM - 1);  // clamped, in-bounds

    const float* ap = aws + ((size_t)b * SDIM + trow) * SDIM + kb;
    const float* bp = x + ((size_t)kb * BDIM + b) * FDIM + fc;
    const int bstep = BDIM * FDIM;      // per +1 in s

    v8f acc = {};
    for (int kk = 0; kk < SDIM; kk += 4) {
        v2f a, bv;
        a.x  = ap[0];
        a.y  = ap[1];
        bv.x = bp[0];
        bv.y = bp[bstep];
        acc = __builtin_amdgcn_wmma_f32_16x16x4_f32(
                false, a, false, bv, (short)0, acc, false, false);
        ap += 4;
        bp += 4 * bstep;
    }

    if (fcol < FDIM) {
        const int rbase = tm * 16 + (lane >> 4) * 8;
        #pragma unroll
        for (int r = 0; r < 8; ++r)
            att[((rbase + r) * BDIM + b) * FDIM + fcol] = acc[r];
    }
}

// ---------------------------------------------------------------------------
// Kernel 5: masked max + mean pool over t, concat [max, mean] -> (B, 2F)
// ---------------------------------------------------------------------------
__global__ __launch_bounds__(256)
void sfe_pool_kernel(const float* __restrict__ att,
                     const int*   __restrict__ lens,
                     float* __restrict__ out) {
    const int id = blockIdx.x * blockDim.x + threadIdx.x;
    if (id >= BDIM * FDIM) return;
    const int b = id / FDIM, f = id % FDIM;
    const int L = lens[b];
    float mx = -INFINITY, sm = 0.f;
    for (int t = 0; t < L; ++t) {
        const float v = att[((size_t)t * BDIM + b) * FDIM + f];
        mx = fmaxf(mx, v);
        sm += v;
    }
    out[b * (2 * FDIM) + f]        = mx;
    out[b * (2 * FDIM) + FDIM + f] = sm / (float)L;
}

// ---------------------------------------------------------------------------
// Launcher
// inputs: [0]=input(S,B,F) f32, [1]=sequence_lengths(B) i32,
//         [2]=Wq(F,H) f32, [3]=Wk(F,H) f32, [4]=b(H) f32, [5]=v(H) f32
// output: (B, 2F) f32.  Workspace: ~10.75 MB of floats.
// ---------------------------------------------------------------------------
extern "C" void kernel_launch(void* const* d_in, const int* in_sizes, int n_in,
                              void* d_out, int out_size, void* d_ws, size_t ws_size,
                              hipStream_t stream) {
    const float* x    = (const float*)d_in[0];
    const int*   lens = (const int*)  d_in[1];
    const float* Wq   = (const float*)d_in[2];
    const float* Wk   = (const float*)d_in[3];
    const float* bias = (const float*)d_in[4];
    const float* vvec = (const float*)d_in[5];
    float* out = (float*)d_out;

    float* ws = (float*)d_ws;
    const size_t QOFF = 0;                                   // S*B*H
    const size_t KOFF = QOFF + (size_t)SDIM * BDIM * HDIM;   // S*B*H
    const size_t EOFF = KOFF + (size_t)SDIM * BDIM * HDIM;   // B*S*S
    const size_t AOFF = EOFF + (size_t)BDIM * SDIM * SDIM;   // S*B*F
    float* qws = ws + QOFF;
    float* kws = ws + KOFF;
    float* ews = ws + EOFF;
    float* att = ws + AOFF;

    // 1) projections (WMMA f32): 4096x200 @ 200x100, for Wq and Wk(+bias)
    sfe_proj_kernel<<<dim3((SDIM * BDIM) / 16, (HDIM + 15) / 16, 2), 32, 0, stream>>>(
        x, Wq, Wk, bias, qws, kws);

    // 2) additive-attention energies + sequence mask
    sfe_energy_kernel<<<dim3(SDIM / 16, SDIM / 16, BDIM), 256, 0, stream>>>(
        qws, kws, vvec, lens, ews);

    // 3) softmax over s (in place)
    sfe_softmax_kernel<<<dim3(BDIM * SDIM), 256, 0, stream>>>(ews);

    // 4) attended = a @ x per batch (WMMA f32)
    sfe_attend_kernel<<<dim3(SDIM / 16, (FDIM + 15) / 16, BDIM), 32, 0, stream>>>(
        ews, x, att);

    // 5) masked max/mean pooling over t, concat
    sfe_pool_kernel<<<dim3((BDIM * FDIM + 255) / 256), 256, 0, stream>>>(
        att, lens, out);
}